// ChunkwiseCausalAttention_19756849562333
// MI455X (gfx1250) — compile-verified
//
#include <hip/hip_runtime.h>
#include <hip/hip_bf16.h>

// ---------------------------------------------------------------------------
// Types / CDNA5 feature helpers
// ---------------------------------------------------------------------------
typedef float v2f __attribute__((ext_vector_type(2)));
typedef float v8f __attribute__((ext_vector_type(8)));
typedef int v4i_vs __attribute__((vector_size(16)));   // matches builtin param

#if __has_builtin(__builtin_amdgcn_global_load_async_to_lds_b128)
#define HAVE_ASYNC_LDS 1
#else
#define HAVE_ASYNC_LDS 0
#endif

#if __has_builtin(__builtin_amdgcn_wmma_f32_16x16x4_f32)
#define HAVE_WMMA_F32 1
#else
#define HAVE_WMMA_F32 0
#endif

// D = A(16x4 f32) * B(4x16 f32) + C(16x16 f32), wave32, exact fp32.
__device__ __forceinline__ v8f wmma_f32(v2f a, v2f b, v8f c) {
#if HAVE_WMMA_F32
  return __builtin_amdgcn_wmma_f32_16x16x4_f32(
      /*neg_a=*/false, a, /*neg_b=*/false, b,
      /*c_mod=*/(short)0, c, /*reuse_a=*/false, /*reuse_b=*/false);
#else
  c[0] += a[0] * b[0] + a[1] * b[1];   // placeholder: keeps TU compiling
  return c;
#endif
}

// 16-byte global -> LDS copy, async on gfx1250.
__device__ __forceinline__ void async_copy16(void* ldst, const void* gsrc) {
#if HAVE_ASYNC_LDS
  __builtin_amdgcn_global_load_async_to_lds_b128(
      (__attribute__((address_space(1))) v4i_vs*)gsrc,
      (__attribute__((address_space(3))) v4i_vs*)ldst, 0, 0);
#else
  *(float4*)ldst = *(const float4*)gsrc;
#endif
}

// Wait until at most N async-to-LDS groups are still in flight.
template <int N>
__device__ __forceinline__ void async_wait_upto() {
#if HAVE_ASYNC_LDS
#if __has_builtin(__builtin_amdgcn_s_wait_asynccnt)
  __builtin_amdgcn_s_wait_asynccnt(N);
#else
  asm volatile("s_wait_asynccnt %0" ::"i"(N) : "memory");
#endif
#endif
}

// ---------------------------------------------------------------------------
// Problem constants
// ---------------------------------------------------------------------------
#define BATCH 2
#define SEQ   2048
#define DMODEL 2048
#define NHEAD 16
#define DHEAD 128
#define NQKV  6144          // 3*H*Dh
#define ROWS  4096          // B*S
#define QK_SCALE 0.08838834764831845f   // 1/sqrt(128)

// GEMM tiling
#define BM 64
#define BN 128
#define BK 16
#define KITER (DMODEL / BK)  // 128
#define LDA_PAD 20   // floats per ldsA row (16 + 4 pad, keeps 16B align)
#define LDB_PAD 132  // floats per ldsB row (128 + 4 pad)

// ---------------------------------------------------------------------------
// Kernel 1: QKV projection.  qkv = x @ Wqkv + bqkv, scattered to
// Q (pre-scaled), K, V in [B, H, S, Dh] fp32 layout.
// 256 threads = 8 waves (2x4); tile 64(M) x 128(N); K step 16;
// double-buffered async-to-LDS pipeline (3 asyncs/wave/iter, wait <= 3).
// ---------------------------------------------------------------------------
__global__ __launch_bounds__(256) void qkv_proj_kernel(
    const float* __restrict__ x, const float* __restrict__ Wqkv,
    const float* __restrict__ bqkv,
    float* __restrict__ Q, float* __restrict__ K, float* __restrict__ V) {
  __shared__ float ldsA[2][BM][LDA_PAD];
  __shared__ float ldsB[2][BK][LDB_PAD];

  const int tid  = threadIdx.x;
  const int lane = tid & 31;
  const int wv   = tid >> 5;
  const int warpM = wv >> 2;       // 0..1
  const int warpN = wv & 3;        // 0..3
  const int half = lane >> 4;      // 0..1
  const int ln16 = lane & 15;      // matrix row (A) / col (B,C)
  const int m0 = blockIdx.y * BM;
  const int n0 = blockIdx.x * BN;

  const int rowA = tid >> 2, c4 = (tid & 3) * 4;     // A stage: 64x16
  const int rowB = tid >> 4, c8 = (tid & 15) * 8;    // B stage: 16x128

  v8f acc[2][2] = {};

  // prologue: stage K-chunk 0 into buffer 0
  async_copy16(&ldsA[0][rowA][c4], &x[(size_t)(m0 + rowA) * DMODEL + c4]);
  {
    const float* g = &Wqkv[(size_t)rowB * NQKV + n0 + c8];
    async_copy16(&ldsB[0][rowB][c8], g);
    async_copy16(&ldsB[0][rowB][c8 + 4], g + 4);
  }

  for (int i = 0; i < KITER; ++i) {
    const int cur = i & 1, nxt = cur ^ 1;
    const bool more = (i + 1) < KITER;
    if (more) {                          // stage chunk i+1 into other buffer
      const int k0n = (i + 1) * BK;
      async_copy16(&ldsA[nxt][rowA][c4],
                   &x[(size_t)(m0 + rowA) * DMODEL + k0n + c4]);
      const float* g = &Wqkv[(size_t)(k0n + rowB) * NQKV + n0 + c8];
      async_copy16(&ldsB[nxt][rowB][c8], g);
      async_copy16(&ldsB[nxt][rowB][c8 + 4], g + 4);
      async_wait_upto<3>();              // chunk i complete (in-order)
    } else {
      async_wait_upto<0>();
    }
    __syncthreads();

#pragma unroll
    for (int s = 0; s < 4; ++s) {        // K = 16 -> 4 wmma steps of 4
      const int kk = 4 * s + 2 * half;
      v2f a[2], b[2];
#pragma unroll
      for (int mt = 0; mt < 2; ++mt) {
        const int m = warpM * 32 + mt * 16 + ln16;
        a[mt] = *(const v2f*)&ldsA[cur][m][kk];
      }
#pragma unroll
      for (int nt = 0; nt < 2; ++nt) {
        const int col = warpN * 32 + nt * 16 + ln16;
        v2f bb;
        bb[0] = ldsB[cur][kk][col];
        bb[1] = ldsB[cur][kk + 1][col];
        b[nt] = bb;
      }
#pragma unroll
      for (int mt = 0; mt < 2; ++mt)
#pragma unroll
        for (int nt = 0; nt < 2; ++nt)
          acc[mt][nt] = wmma_f32(a[mt], b[nt], acc[mt][nt]);
    }
    __syncthreads();
  }

  // Epilogue: bias + scatter to Q/K/V [B,H,S,Dh]; Q pre-scaled by 1/sqrt(Dh)
#pragma unroll
  for (int mt = 0; mt < 2; ++mt) {
#pragma unroll
    for (int nt = 0; nt < 2; ++nt) {
      const int gj = n0 + warpN * 32 + nt * 16 + ln16;   // 0..6143
      const int which = gj >> 11;                        // 0=q 1=k 2=v
      const int rem = gj & 2047;
      const int h = rem >> 7, d = rem & 127;
      const float bias = bqkv[gj];
#pragma unroll
      for (int r = 0; r < 8; ++r) {
        const int gi = m0 + warpM * 32 + mt * 16 + r + 8 * half;  // 0..4095
        const int bb = gi >> 11, sg = gi & 2047;
        const size_t off = ((size_t)(bb * NHEAD + h) * SEQ + sg) * DHEAD + d;
        const float vv = acc[mt][nt][r] + bias;
        if (which == 0)      Q[off] = vv * QK_SCALE;
        else if (which == 1) K[off] = vv;
        else                 V[off] = vv;
      }
    }
  }
}

// ---------------------------------------------------------------------------
// Kernel 2: flash-style causal attention, fp32 WMMA.
// Grid (S/128, B*H); block 256 = 8 waves; each wave owns 16 query rows with
// the whole Dh=128 Q fragment in registers. 16-key chunks double-buffered:
// V via async-to-LDS (wait <= 2), K via register prefetch + LDS transpose.
// ---------------------------------------------------------------------------
__global__ __launch_bounds__(256) void attn_kernel(
    const float* __restrict__ Q, const float* __restrict__ K,
    const float* __restrict__ V, float* __restrict__ O) {
  __shared__ float ldsKT[2][DHEAD][17];   // [buf][dim][key], padded
  __shared__ float ldsV[2][16][LDB_PAD];  // [buf][key][dim], padded
  __shared__ float ldsP[8][16][18];       // per-wave P restage, padded

  const int tid  = threadIdx.x;
  const int lane = tid & 31;
  const int wv   = tid >> 5;
  const int half = lane >> 4;
  const int ln16 = lane & 15;
  const int bh = blockIdx.y;                 // 0..31
  const int b  = bh >> 4, h = bh & 15;
  const int qt = blockIdx.x;                 // 0..15
  const int qbase = qt * 128 + wv * 16;      // first query row of this wave
  const size_t headoff = (size_t)bh * SEQ * DHEAD;

  const int key = tid >> 4;                  // staging: 0..15
  const int d0  = (tid & 15) * 8;            // staging: 0..120

  // Q fragment: 16 rows x 128 dims, A-operand layout (32 x v2f)
  v2f qf[32];
  {
    const float* qrow = &Q[headoff + (size_t)(qbase + ln16) * DHEAD];
#pragma unroll
    for (int s = 0; s < 32; ++s)
      qf[s] = *(const v2f*)&qrow[4 * s + 2 * half];
  }

  v8f acc[8] = {};
  float m_run[8], l_run[8];
#pragma unroll
  for (int r = 0; r < 8; ++r) { m_run[r] = -1e30f; l_run[r] = 0.0f; }

  const int nchunks = qt * 8 + 8;            // keys 0 .. qt*128+127

  // prologue: prefetch K chunk 0 into registers, async V chunk 0 into buf 0
  float4 kreg0, kreg1;
  {
    const size_t base = headoff + (size_t)key * DHEAD + d0;
    kreg0 = *(const float4*)&K[base];
    kreg1 = *(const float4*)&K[base + 4];
    async_copy16(&ldsV[0][key][d0], &V[base]);
    async_copy16(&ldsV[0][key][d0 + 4], &V[base + 4]);
  }

  for (int c = 0; c < nchunks; ++c) {
    const int cur = c & 1, nxt = cur ^ 1;
    // transpose prefetched K (chunk c) into [dim][key]
    ldsKT[cur][d0 + 0][key] = kreg0.x; ldsKT[cur][d0 + 1][key] = kreg0.y;
    ldsKT[cur][d0 + 2][key] = kreg0.z; ldsKT[cur][d0 + 3][key] = kreg0.w;
    ldsKT[cur][d0 + 4][key] = kreg1.x; ldsKT[cur][d0 + 5][key] = kreg1.y;
    ldsKT[cur][d0 + 6][key] = kreg1.z; ldsKT[cur][d0 + 7][key] = kreg1.w;

    const bool more = (c + 1) < nchunks;
    if (more) {                              // stage chunk c+1
      const size_t base = headoff + (size_t)((c + 1) * 16 + key) * DHEAD + d0;
      kreg0 = *(const float4*)&K[base];
      kreg1 = *(const float4*)&K[base + 4];
      async_copy16(&ldsV[nxt][key][d0], &V[base]);
      async_copy16(&ldsV[nxt][key][d0 + 4], &V[base + 4]);
      async_wait_upto<2>();                  // V chunk c complete (in-order)
    } else {
      async_wait_upto<0>();
    }
    __syncthreads();

    const bool active = (c * 16) <= (qbase + 15);   // wave-uniform
    if (active) {
      // scores = Q * K^T  (16x16, contraction over Dh=128 -> 32 wmmas)
      v8f sc = {};
#pragma unroll
      for (int s = 0; s < 32; ++s) {
        const int kk = 4 * s + 2 * half;
        v2f bb;
        bb[0] = ldsKT[cur][kk][ln16];
        bb[1] = ldsKT[cur][kk + 1][ln16];
        sc = wmma_f32(qf[s], bb, sc);
      }
      // ---- causal mask + online softmax ----
      const int key_g = c * 16 + ln16;
      float p[8];
#pragma unroll
      for (int r = 0; r < 8; ++r) {
        const int qg = qbase + r + 8 * half;
        float sv = (key_g <= qg) ? sc[r] : -1e30f;
        float mx = sv;
#pragma unroll
        for (int off = 8; off >= 1; off >>= 1)
          mx = fmaxf(mx, __shfl_xor(mx, off, 32));
        const float m_new = fmaxf(m_run[r], mx);
        const float alpha = __expf(m_run[r] - m_new);
        const float pv = __expf(sv - m_new);
        float rs = pv;
#pragma unroll
        for (int off = 8; off >= 1; off >>= 1)
          rs += __shfl_xor(rs, off, 32);
        l_run[r] = l_run[r] * alpha + rs;
        m_run[r] = m_new;
        p[r] = pv;
#pragma unroll
        for (int t = 0; t < 8; ++t) acc[t][r] *= alpha;
      }
      // restage P through per-wave LDS into A-operand layout
#pragma unroll
      for (int r = 0; r < 8; ++r)
        ldsP[wv][r + 8 * half][ln16] = p[r];
      // out += P(16x16) * V(16x128): 4 k-steps x 8 N-tiles
#pragma unroll
      for (int s4 = 0; s4 < 4; ++s4) {
        const int kk = 4 * s4 + 2 * half;
        const v2f a = *(const v2f*)&ldsP[wv][ln16][kk];
#pragma unroll
        for (int t = 0; t < 8; ++t) {
          v2f bb;
          bb[0] = ldsV[cur][kk][t * 16 + ln16];
          bb[1] = ldsV[cur][kk + 1][t * 16 + ln16];
          acc[t] = wmma_f32(a, bb, acc[t]);
        }
      }
    }
    __syncthreads();
  }

  // epilogue: normalize and write O as [B, S, H*Dh]
#pragma unroll
  for (int t = 0; t < 8; ++t) {
#pragma unroll
    for (int r = 0; r < 8; ++r) {
      const int sg = qbase + r + 8 * half;
      const float ov = acc[t][r] / l_run[r];
      O[((size_t)(b * SEQ + sg)) * DMODEL + h * DHEAD + t * 16 + ln16] = ov;
    }
  }
}

// ---------------------------------------------------------------------------
// Kernel 3: output projection.  out = O @ Wout + bout.  Same pipeline as K1.
// ---------------------------------------------------------------------------
__global__ __launch_bounds__(256) void out_proj_kernel(
    const float* __restrict__ A, const float* __restrict__ Wout,
    const float* __restrict__ bout, float* __restrict__ out) {
  __shared__ float ldsA[2][BM][LDA_PAD];
  __shared__ float ldsB[2][BK][LDB_PAD];

  const int tid  = threadIdx.x;
  const int lane = tid & 31;
  const int wv   = tid >> 5;
  const int warpM = wv >> 2, warpN = wv & 3;
  const int half = lane >> 4, ln16 = lane & 15;
  const int m0 = blockIdx.y * BM;
  const int n0 = blockIdx.x * BN;

  const int rowA = tid >> 2, c4 = (tid & 3) * 4;
  const int rowB = tid >> 4, c8 = (tid & 15) * 8;

  v8f acc[2][2] = {};

  async_copy16(&ldsA[0][rowA][c4], &A[(size_t)(m0 + rowA) * DMODEL + c4]);
  {
    const float* g = &Wout[(size_t)rowB * DMODEL + n0 + c8];
    async_copy16(&ldsB[0][rowB][c8], g);
    async_copy16(&ldsB[0][rowB][c8 + 4], g + 4);
  }

  for (int i = 0; i < KITER; ++i) {
    const int cur = i & 1, nxt = cur ^ 1;
    const bool more = (i + 1) < KITER;
    if (more) {
      const int k0n = (i + 1) * BK;
      async_copy16(&ldsA[nxt][rowA][c4],
                   &A[(size_t)(m0 + rowA) * DMODEL + k0n + c4]);
      const float* g = &Wout[(size_t)(k0n + rowB) * DMODEL + n0 + c8];
      async_copy16(&ldsB[nxt][rowB][c8], g);
      async_copy16(&ldsB[nxt][rowB][c8 + 4], g + 4);
      async_wait_upto<3>();
    } else {
      async_wait_upto<0>();
    }
    __syncthreads();

#pragma unroll
    for (int s = 0; s < 4; ++s) {
      const int kk = 4 * s + 2 * half;
      v2f a[2], bvec[2];
#pragma unroll
      for (int mt = 0; mt < 2; ++mt) {
        const int m = warpM * 32 + mt * 16 + ln16;
        a[mt] = *(const v2f*)&ldsA[cur][m][kk];
      }
#pragma unroll
      for (int nt = 0; nt < 2; ++nt) {
        const int col = warpN * 32 + nt * 16 + ln16;
        v2f bb;
        bb[0] = ldsB[cur][kk][col];
        bb[1] = ldsB[cur][kk + 1][col];
        bvec[nt] = bb;
      }
#pragma unroll
      for (int mt = 0; mt < 2; ++mt)
#pragma unroll
        for (int nt = 0; nt < 2; ++nt)
          acc[mt][nt] = wmma_f32(a[mt], bvec[nt], acc[mt][nt]);
    }
    __syncthreads();
  }

#pragma unroll
  for (int mt = 0; mt < 2; ++mt) {
#pragma unroll
    for (int nt = 0; nt < 2; ++nt) {
      const int gj = n0 + warpN * 32 + nt * 16 + ln16;
      const float bias = bout[gj];
#pragma unroll
      for (int r = 0; r < 8; ++r) {
        const int gi = m0 + warpM * 32 + mt * 16 + r + 8 * half;
        out[(size_t)gi * DMODEL + gj] = acc[mt][nt][r] + bias;
      }
    }
  }
}

// ---------------------------------------------------------------------------
// Host launcher
// ---------------------------------------------------------------------------
extern "C" void kernel_launch(void* const* d_in, const int* in_sizes, int n_in,
                              void* d_out, int out_size, void* d_ws,
                              size_t ws_size, hipStream_t stream) {
  (void)in_sizes; (void)n_in; (void)out_size; (void)ws_size;
  const float* x    = (const float*)d_in[0];
  const float* Wqkv = (const float*)d_in[1];
  const float* bqkv = (const float*)d_in[2];
  const float* Wout = (const float*)d_in[3];
  const float* bout = (const float*)d_in[4];
  float* out = (float*)d_out;

  const size_t per_buf = (size_t)BATCH * NHEAD * SEQ * DHEAD;  // 8.39M floats
  float* Qb = (float*)d_ws;
  float* Kb = Qb + per_buf;
  float* Vb = Kb + per_buf;
  float* Ob = Vb + per_buf;          // [B, S, H*Dh] = 8.39M floats

  qkv_proj_kernel<<<dim3(NQKV / BN, ROWS / BM), 256, 0, stream>>>(
      x, Wqkv, bqkv, Qb, Kb, Vb);
  attn_kernel<<<dim3(SEQ / 128, BATCH * NHEAD), 256, 0, stream>>>(
      Qb, Kb, Vb, Ob);
  out_proj_kernel<<<dim3(DMODEL / BN, ROWS / BM), 256, 0, stream>>>(
      Ob, Wout, bout, out);
}